// Block_71554155151855
// MI455X (gfx1250) — compile-verified
//
#include <hip/hip_runtime.h>
#include <math.h>

// ---------------------------------------------------------------------------
// CDNA5 / gfx1250: wave32. All shared-weight linear maps run on
// V_WMMA_F32_16X16X4_F32 with LDS-staged A/B tiles (ds_load + v_wmma inner
// loop, uniform EXEC). Per-edge equivariant plumbing stays on VALU.
// ---------------------------------------------------------------------------

typedef __attribute__((ext_vector_type(2))) float v2f;
typedef __attribute__((ext_vector_type(8))) float v8f;

#define NNODES 10000
#define NEDGES 160000

#define KT   48          // K-chunk staged in LDS
#define KSTR (KT + 2)    // padded row stride (8B aligned, spreads banks)

__device__ __forceinline__ float siluf(float x) { return x / (1.f + __expf(-x)); }

__device__ __forceinline__ float wred(float v) {
  for (int off = 16; off > 0; off >>= 1) v += __shfl_down(v, off);
  return __shfl(v, 0);
}

// ---------------------------------------------------------------------------
// Generic WMMA GEMM over edge tiles.
//   Y[e, (yoff+z) + o*osy] (+)= bias[o] + sum_c W[o*ldw + woff + c] *
//                               X[row(e)*ldx + (xoff+z) + c*csx]
// row(e) = gather ? gather[e] : e.
// grid = (ceil(E/64), ceil(O/16), nm), block = 128 (4 waves).
// Each block: one 16-wide output tile, 4 edge tiles of 16 (one per wave).
// W tile staged once per block into LDS; X tile staged per wave.
// A fragment (16x4 f32): lanes 0-15 hold K=k,k+1; lanes 16-31 hold K=k+2,k+3.
// D fragment (16x16 f32): VGPR i, lanes 0-15 -> row i, lanes 16-31 -> row i+8.
// ---------------------------------------------------------------------------
__global__ void __launch_bounds__(128)
k_gemm(const float* __restrict__ X, int ldx, int csx, int xoff,
       const float* __restrict__ W, int ldw, int woff,
       const float* __restrict__ bias,
       float* __restrict__ Y, int ldy, int osy, int yoff,
       int E, int O, int K, int accum,
       const int* __restrict__ gather) {
  __shared__ float Wl[16 * KSTR];
  __shared__ float Xl[4][16 * KSTR];

  const int tid  = threadIdx.x;
  const int wv   = tid >> 5;            // wave id 0..3
  const int lane = tid & 31;
  const int e0   = blockIdx.x * 64 + wv * 16;
  const int o0   = blockIdx.y * 16;
  const int z    = blockIdx.z;          // m-component

  const int r  = lane & 15;
  const int kb = (lane >> 4) << 1;      // 0 or 2

  float* Xw = Xl[wv];
  v8f acc = {0.f, 0.f, 0.f, 0.f, 0.f, 0.f, 0.f, 0.f};

  for (int kt = 0; kt < K; kt += KT) {
    const int chunk = (K - kt < KT) ? (K - kt) : KT;
    __syncthreads();
    // stage W tile (block-cooperative): wave wv takes rows wv, wv+4, wv+8, wv+12
    for (int n = wv; n < 16; n += 4) {
      const int wr = (o0 + n < O) ? (o0 + n) : (O - 1);   // clamp, store masked later
      const float* wp = W + (size_t)wr * ldw + woff + kt;
      for (int k = lane; k < chunk; k += 32) Wl[n * KSTR + k] = wp[k];
    }
    // stage this wave's X tile (16 edge rows, gather + clamp)
    for (int n = 0; n < 16; ++n) {
      const int er = (e0 + n < E) ? (e0 + n) : (E - 1);
      const int xr = gather ? gather[er] : er;
      const float* xp = X + (size_t)xr * ldx + (xoff + z) + (size_t)kt * csx;
      for (int k = lane; k < chunk; k += 32) Xw[n * KSTR + k] = xp[(size_t)k * csx];
    }
    // prefetch next W chunk into cache while this one computes
    if (kt + KT < K) {
      const int wr = (o0 + r < O) ? (o0 + r) : (O - 1);
      __builtin_prefetch(W + (size_t)wr * ldw + woff + kt + KT, 0, 1);
    }
    __syncthreads();
    // uniform-EXEC WMMA loop: ds_load pairs + v_wmma
    for (int k = 0; k < chunk; k += 4) {
      v2f a, b;
      a.x = Xw[r * KSTR + k + kb];
      a.y = Xw[r * KSTR + k + kb + 1];
      b.x = Wl[r * KSTR + k + kb];
      b.y = Wl[r * KSTR + k + kb + 1];
      acc = __builtin_amdgcn_wmma_f32_16x16x4_f32(false, a, false, b,
                                                  (short)0, acc, false, false);
    }
  }

  if (o0 + r >= O) return;              // mask padded output columns
  const float bv  = bias ? bias[o0 + r] : 0.f;
  const int   rhi = (lane >> 4) * 8;
#pragma unroll
  for (int i = 0; i < 8; ++i) {
    const int mr = e0 + rhi + i;
    if (mr >= E) continue;              // mask padded edge rows
    const size_t yi = (size_t)mr * ldy + (size_t)(o0 + r) * osy + (yoff + z);
    const float v = acc[i] + bv;
    if (accum) Y[yi] += v; else Y[yi] = v;
  }
}

// --------------------------- equivariant layernorm -------------------------
__global__ void k_equiv_ln(const float* __restrict__ X, float* __restrict__ Y,
                           const float* __restrict__ w0, const float* __restrict__ b0,
                           const float* __restrict__ w1, const float* __restrict__ w2) {
  const size_t row = blockIdx.x;
  const int lane = threadIdx.x;
  const float* x = X + row * 360;
  float* y = Y + row * 360;

  float s0[3]; float sum = 0.f;
#pragma unroll
  for (int i = 0; i < 3; i++) { s0[i] = x[lane + 32 * i]; sum += s0[i]; }
  const float mean = wred(sum) * (1.f / 96.f);
  float sq = 0.f;
#pragma unroll
  for (int i = 0; i < 3; i++) { float d = s0[i] - mean; sq += d * d; }
  const float inv = rsqrtf(wred(sq) * (1.f / 96.f) + 1e-5f);
#pragma unroll
  for (int i = 0; i < 3; i++) {
    int j = lane + 32 * i;
    y[j] = (s0[i] - mean) * inv * w0[j] + b0[j];
  }

  float v[5]; float ss = 0.f;
#pragma unroll
  for (int i = 0; i < 5; i++) {
    int j = lane + 32 * i;
    float t = (j < 144) ? x[96 + j] : 0.f;
    v[i] = t; ss += t * t;
  }
  const float inv1 = rsqrtf(wred(ss) * (1.f / 48.f) + 1e-5f);
#pragma unroll
  for (int i = 0; i < 5; i++) {
    int j = lane + 32 * i;
    if (j < 144) y[96 + j] = v[i] * inv1 * w1[j / 3];
  }

  float u[4]; float ss2 = 0.f;
#pragma unroll
  for (int i = 0; i < 4; i++) {
    int j = lane + 32 * i;
    float t = (j < 120) ? x[240 + j] : 0.f;
    u[i] = t; ss2 += t * t;
  }
  const float inv2 = rsqrtf(wred(ss2) * (1.f / 24.f) + 1e-5f);
#pragma unroll
  for (int i = 0; i < 4; i++) {
    int j = lane + 32 * i;
    if (j < 120) y[240 + j] = u[i] * inv2 * w2[j / 5];
  }
}

// LN over 64 cols + SiLU, in place (eps 1e-6). One wave per row.
__global__ void k_ln_silu(float* __restrict__ X, const float* __restrict__ w,
                          const float* __restrict__ b) {
  const size_t row = blockIdx.x;
  const int lane = threadIdx.x;
  float* x = X + row * 64;
  float a0 = x[lane], a1 = x[lane + 32];
  const float mean = wred(a0 + a1) * (1.f / 64.f);
  const float d0 = a0 - mean, d1 = a1 - mean;
  const float inv = rsqrtf(wred(d0 * d0 + d1 * d1) * (1.f / 64.f) + 1e-6f);
  float r0 = d0 * inv * w[lane] + b[lane];
  float r1 = d1 * inv * w[lane + 32] + b[lane + 32];
  x[lane]      = siluf(r0);
  x[lane + 32] = siluf(r1);
}

__global__ void k_silu_range(float* __restrict__ X, int ld, int off, int w, long n) {
  long i = (long)blockIdx.x * blockDim.x + threadIdx.x;
  if (i >= n) return;
  long r = i / w; int c = (int)(i % w);
  float* p = X + r * ld + off + c;
  *p = siluf(*p);
}

// Wigner rotation: fwd: Y[e,c,n]=sum_m D[e,n,m]X[e,c,m]; trans: D[e,m,n].
__global__ void k_rot(const float* __restrict__ X, float* __restrict__ Y,
                      const float* __restrict__ D, int off, int C, int M,
                      int trans, int E) {
  int idx = blockIdx.x * blockDim.x + threadIdx.x;
  if (idx >= E * C) return;
  int e = idx / C, c = idx % C;
  const float* x = X + (size_t)e * 360 + off + c * M;
  float*       y = Y + (size_t)e * 360 + off + c * M;
  const float* d = D + (size_t)e * M * M;
  float in[5];
  for (int m = 0; m < M; m++) in[m] = x[m];
  for (int nn = 0; nn < M; nn++) {
    float acc = 0.f;
    for (int m = 0; m < M; m++)
      acc += (trans ? d[m * M + nn] : d[nn * M + m]) * in[m];
    y[nn] = acc;
  }
}

__global__ void k_scale_radial(float* __restrict__ X, const float* __restrict__ R, int E) {
  long i = (long)blockIdx.x * blockDim.x + threadIdx.x;
  if (i >= (long)E * 360) return;
  int e = (int)(i / 360), j = (int)(i % 360);
  int ri = (j < 96) ? j : (j < 240) ? 96 + (j - 96) / 3 : 144 + (j - 240) / 5;
  X[i] *= R[(size_t)e * 168 + ri];
}

__global__ void k_qk_score(const float* __restrict__ q, const float* __restrict__ k,
                           float* __restrict__ alpha, int E) {
  int i = blockIdx.x * blockDim.x + threadIdx.x;
  if (i >= E * 4) return;
  int e = i >> 2, h = i & 3;
  const float* qp = q + (size_t)e * 64 + h * 16;
  const float* kp = k + (size_t)e * 64 + h * 16;
  float s = 0.f;
#pragma unroll
  for (int j = 0; j < 16; j++) s += qp[j] * kp[j];
  alpha[i] = s * 0.25f;
}

__global__ void k_onsite(float* __restrict__ value, const float* __restrict__ lin,
                         const float* __restrict__ evec, int E) {
  long i = (long)blockIdx.x * blockDim.x + threadIdx.x;
  if (i >= (long)E * 360) return;
  int e = (int)(i / 360);
  float x = evec[e * 3], y = evec[e * 3 + 1], z = evec[e * 3 + 2];
  if (x * x + y * y + z * z < 1e-20f) value[i] = lin[i];
}

// heads2vec(vec2heads(v) * alpha) == v * alpha[head(j)]
__global__ void k_alpha_scale(const float* __restrict__ v, const float* __restrict__ alpha,
                              float* __restrict__ out, int E) {
  long i = (long)blockIdx.x * blockDim.x + threadIdx.x;
  if (i >= (long)E * 360) return;
  int e = (int)(i / 360), j = (int)(i % 360);
  int h = (j < 96) ? j / 24 : (j < 240) ? (j - 96) / 36 : (j - 240) / 30;
  out[i] = v[i] * alpha[(size_t)e * 4 + h];
}

__global__ void k_segsum(const float* __restrict__ msg, const int* __restrict__ dst,
                         float* __restrict__ node, int E) {
  long i = (long)blockIdx.x * blockDim.x + threadIdx.x;
  if (i >= (long)E * 360) return;
  int e = (int)(i / 360), j = (int)(i % 360);
  atomicAdd(&node[(size_t)dst[e] * 360 + j], msg[i]);
}

__global__ void k_zero(float* __restrict__ p, long n) {
  long i = (long)blockIdx.x * blockDim.x + threadIdx.x;
  if (i < n) p[i] = 0.f;
}

__global__ void k_copy(float* __restrict__ dst, const float* __restrict__ src,
                       int ld, int off, int w, long n) {
  long i = (long)blockIdx.x * blockDim.x + threadIdx.x;
  if (i >= n) return;
  long r = i / w; int c = (int)(i % w);
  dst[r * ld + off + c] = src[r * ld + off + c];
}

// ---------------------------------------------------------------------------
static inline unsigned gdiv(long n, int b) { return (unsigned)((n + b - 1) / b); }

static void gemm(hipStream_t st, const float* X, int ldx, int csx, int xoff,
                 const float* W, int ldw, int woff, const float* bias,
                 float* Y, int ldy, int osy, int yoff,
                 int E, int O, int K, int accum, const int* gather, int nm) {
  dim3 g(gdiv(E, 64), (O + 15) / 16, nm);
  k_gemm<<<g, dim3(128), 0, st>>>(X, ldx, csx, xoff, W, ldw, woff, bias,
                                  Y, ldy, osy, yoff, E, O, K, accum, gather);
}

static void irreps(hipStream_t st, const float* X, const float* W0,
                   const float* W1, const float* W2, float* Y, int E, int accum) {
  gemm(st, X, 360, 1, 0,   W0, 96, 0, nullptr, Y, 360, 1, 0,   E, 96, 96, accum, nullptr, 1);
  gemm(st, X, 360, 3, 96,  W1, 48, 0, nullptr, Y, 360, 3, 96,  E, 48, 48, accum, nullptr, 3);
  gemm(st, X, 360, 5, 240, W2, 24, 0, nullptr, Y, 360, 5, 240, E, 24, 24, accum, nullptr, 5);
}

extern "C" void kernel_launch(void* const* d_in, const int* in_sizes, int n_in,
                              void* d_out, int out_size, void* d_ws, size_t ws_size,
                              hipStream_t stream) {
  (void)in_sizes; (void)n_in; (void)out_size; (void)ws_size;
  const int NN = NNODES, NE = NEDGES;

  const float* node_fea = (const float*)d_in[0];
  const float* edge_fea = (const float*)d_in[1];
  const float* ele      = (const float*)d_in[3];   // edge_length_embedding
  const float* edge_vec = (const float*)d_in[4];
  const float* D1 = (const float*)d_in[5];
  const float* D2 = (const float*)d_in[6];
  const float* Wl0 = (const float*)d_in[7];
  const float* Wl1 = (const float*)d_in[8];
  const float* Wl2 = (const float*)d_in[9];
  const float* preW0 = (const float*)d_in[10];
  const float* preW1 = (const float*)d_in[11];
  const float* preW2 = (const float*)d_in[12];
  const float* ln_w0 = (const float*)d_in[13];
  const float* ln_b0 = (const float*)d_in[14];
  const float* ln_w1 = (const float*)d_in[15];
  const float* ln_w2 = (const float*)d_in[16];
  const float* qk_W1 = (const float*)d_in[17];
  const float* qk_b1 = (const float*)d_in[18];
  const float* qk_lnw = (const float*)d_in[19];
  const float* qk_lnb = (const float*)d_in[20];
  const float* qk_W2 = (const float*)d_in[21];
  const float* qk_b2 = (const float*)d_in[22];
  const float* a_W1 = (const float*)d_in[23];
  const float* a_b1 = (const float*)d_in[24];
  const float* a_ln1w = (const float*)d_in[25];
  const float* a_ln1b = (const float*)d_in[26];
  const float* a_W2 = (const float*)d_in[27];
  const float* a_b2 = (const float*)d_in[28];
  const float* a_ln2w = (const float*)d_in[29];
  const float* a_ln2b = (const float*)d_in[30];
  const float* a_W3 = (const float*)d_in[31];
  const float* a_b3 = (const float*)d_in[32];
  const float* r_W1 = (const float*)d_in[33];
  const float* r_b1 = (const float*)d_in[34];
  const float* r_W2 = (const float*)d_in[35];
  const float* r_b2 = (const float*)d_in[36];
  const int* edge_src = (const int*)d_in[37];
  const int* edge_dst = (const int*)d_in[38];

  float* node_out = (float*)d_out;
  float* edge_out = node_out + (size_t)NN * 360;

  // --- workspace carve-up (floats) ---
  float* w = (float*)d_ws;
  size_t off = 0;
  auto alloc = [&](size_t n) { float* p = w + off; off += n; return p; };
  float* node_n   = alloc((size_t)NN * 360);
  float* node_msg = alloc((size_t)NN * 360);
  float* edge_n   = alloc((size_t)NE * 360);
  float* message  = alloc((size_t)NE * 360);
  float* tmpA     = alloc((size_t)NE * 360);
  float* tmpB     = alloc((size_t)NE * 360);
  float* value    = alloc((size_t)NE * 360);
  float* radial   = alloc((size_t)NE * 168);
  float* qb       = alloc((size_t)NE * 64);
  float* kb       = alloc((size_t)NE * 64);
  float* t1       = alloc((size_t)NE * 64);
  float* alpha    = alloc((size_t)NE * 4);

  const long EW = (long)NE * 360;

  // 1) equivariant layernorms
  k_equiv_ln<<<NN, 32, 0, stream>>>(node_fea, node_n, ln_w0, ln_b0, ln_w1, ln_w2);
  k_equiv_ln<<<NE, 32, 0, stream>>>(edge_fea, edge_n, ln_w0 + 96, ln_b0 + 96,
                                    ln_w1 + 48, ln_w2 + 24);

  // 2) q = mlp2(node_scalar[dst]), k = mlp2(node_scalar[src])
  gemm(stream, node_n, 360, 1, 0, qk_W1, 96, 0, qk_b1, t1, 64, 1, 0, NE, 64, 96, 0, edge_dst, 1);
  k_ln_silu<<<NE, 32, 0, stream>>>(t1, qk_lnw, qk_lnb);
  gemm(stream, t1, 64, 1, 0, qk_W2, 64, 0, qk_b2, qb, 64, 1, 0, NE, 64, 64, 0, nullptr, 1);

  gemm(stream, node_n, 360, 1, 0, qk_W1 + 64 * 96, 96, 0, qk_b1 + 64, t1, 64, 1, 0, NE, 64, 96, 0, edge_src, 1);
  k_ln_silu<<<NE, 32, 0, stream>>>(t1, qk_lnw + 64, qk_lnb + 64);
  gemm(stream, t1, 64, 1, 0, qk_W2 + 64 * 64, 64, 0, qk_b2 + 64, kb, 64, 1, 0, NE, 64, 64, 0, nullptr, 1);

  k_qk_score<<<gdiv((long)NE * 4, 256), 256, 0, stream>>>(qb, kb, alpha, NE);

  // 3) attention-bias MLP -> alpha += h @ a_W3^T + a_b3
  gemm(stream, ele, 64, 1, 0, a_W1, 64, 0, a_b1, t1, 64, 1, 0, NE, 64, 64, 0, nullptr, 1);
  k_ln_silu<<<NE, 32, 0, stream>>>(t1, a_ln1w, a_ln1b);
  gemm(stream, t1, 64, 1, 0, a_W2, 64, 0, a_b2, qb, 64, 1, 0, NE, 64, 64, 0, nullptr, 1);
  k_ln_silu<<<NE, 32, 0, stream>>>(qb, a_ln2w, a_ln2b);
  gemm(stream, qb, 64, 1, 0, a_W3, 64, 0, a_b3, alpha, 4, 1, 0, NE, 4, 64, 1, nullptr, 1);

  // 4) radial = silu(ele @ r_W1^T + b) @ r_W2^T + b
  gemm(stream, ele, 64, 1, 0, r_W1, 64, 0, r_b1, t1, 64, 1, 0, NE, 64, 64, 0, nullptr, 1);
  k_silu_range<<<gdiv((long)NE * 64, 256), 256, 0, stream>>>(t1, 64, 0, 64, (long)NE * 64);
  gemm(stream, t1, 64, 1, 0, r_W2, 64, 0, r_b2, radial, 168, 1, 0, NE, 168, 64, 0, nullptr, 1);

  // 5) message = pre-linear over concat(src, dst, edge) (weight-column blocks)
  gemm(stream, node_n, 360, 1, 0, preW0, 288, 0,   nullptr, message, 360, 1, 0, NE, 96, 96, 0, edge_src, 1);
  gemm(stream, node_n, 360, 1, 0, preW0, 288, 96,  nullptr, message, 360, 1, 0, NE, 96, 96, 1, edge_dst, 1);
  gemm(stream, edge_n, 360, 1, 0, preW0, 288, 192, nullptr, message, 360, 1, 0, NE, 96, 96, 1, nullptr, 1);
  gemm(stream, node_n, 360, 3, 96, preW1, 144, 0,  nullptr, message, 360, 3, 96, NE, 48, 48, 0, edge_src, 3);
  gemm(stream, node_n, 360, 3, 96, preW1, 144, 48, nullptr, message, 360, 3, 96, NE, 48, 48, 1, edge_dst, 3);
  gemm(stream, edge_n, 360, 3, 96, preW1, 144, 96, nullptr, message, 360, 3, 96, NE, 48, 48, 1, nullptr, 3);
  gemm(stream, node_n, 360, 5, 240, preW2, 72, 0,  nullptr, message, 360, 5, 240, NE, 24, 24, 0, edge_src, 5);
  gemm(stream, node_n, 360, 5, 240, preW2, 72, 24, nullptr, message, 360, 5, 240, NE, 24, 24, 1, edge_dst, 5);
  gemm(stream, edge_n, 360, 5, 240, preW2, 72, 48, nullptr, message, 360, 5, 240, NE, 24, 24, 1, nullptr, 5);

  // 6) so2_conv #1 (radial, equiv_ln[2], silu on scalars)
  k_rot<<<gdiv((long)NE * 48, 256), 256, 0, stream>>>(message, tmpA, D1, 96, 48, 3, 0, NE);
  k_rot<<<gdiv((long)NE * 24, 256), 256, 0, stream>>>(message, tmpA, D2, 240, 24, 5, 0, NE);
  gemm(stream, message, 360, 1, 0, Wl0, 96, 0, nullptr, tmpB, 360, 1, 0, NE, 96, 96, 0, nullptr, 1);
  gemm(stream, tmpA, 360, 3, 96,  Wl1, 48, 0, nullptr, tmpB, 360, 3, 96,  NE, 48, 48, 0, nullptr, 3);
  gemm(stream, tmpA, 360, 5, 240, Wl2, 24, 0, nullptr, tmpB, 360, 5, 240, NE, 24, 24, 0, nullptr, 5);
  k_scale_radial<<<gdiv(EW, 256), 256, 0, stream>>>(tmpB, radial, NE);
  k_equiv_ln<<<NE, 32, 0, stream>>>(tmpB, tmpB, ln_w0 + 192, ln_b0 + 192, ln_w1 + 96, ln_w2 + 48);
  k_silu_range<<<gdiv((long)NE * 96, 256), 256, 0, stream>>>(tmpB, 360, 0, 96, (long)NE * 96);
  k_rot<<<gdiv((long)NE * 48, 256), 256, 0, stream>>>(tmpB, value, D1, 96, 48, 3, 1, NE);
  k_rot<<<gdiv((long)NE * 24, 256), 256, 0, stream>>>(tmpB, value, D2, 240, 24, 5, 1, NE);
  k_copy<<<gdiv((long)NE * 96, 256), 256, 0, stream>>>(value, tmpB, 360, 0, 96, (long)NE * 96);

  // 7) so2_conv #2 (plain)
  k_rot<<<gdiv((long)NE * 48, 256), 256, 0, stream>>>(value, tmpA, D1, 96, 48, 3, 0, NE);
  k_rot<<<gdiv((long)NE * 24, 256), 256, 0, stream>>>(value, tmpA, D2, 240, 24, 5, 0, NE);
  gemm(stream, value, 360, 1, 0, Wl0 + 96 * 96, 96, 0, nullptr, tmpB, 360, 1, 0, NE, 96, 96, 0, nullptr, 1);
  gemm(stream, tmpA, 360, 3, 96,  Wl1 + 48 * 48, 48, 0, nullptr, tmpB, 360, 3, 96,  NE, 48, 48, 0, nullptr, 3);
  gemm(stream, tmpA, 360, 5, 240, Wl2 + 24 * 24, 24, 0, nullptr, tmpB, 360, 5, 240, NE, 24, 24, 0, nullptr, 5);
  k_rot<<<gdiv((long)NE * 48, 256), 256, 0, stream>>>(tmpB, value, D1, 96, 48, 3, 1, NE);
  k_rot<<<gdiv((long)NE * 24, 256), 256, 0, stream>>>(tmpB, value, D2, 240, 24, 5, 1, NE);
  k_copy<<<gdiv((long)NE * 96, 256), 256, 0, stream>>>(value, tmpB, 360, 0, 96, (long)NE * 96);

  // 8) onsite override: value[e] = irreps_lin(message, Wl[2]) where |edge_vec|<1e-10
  irreps(stream, message, Wl0 + 2 * 96 * 96, Wl1 + 2 * 48 * 48, Wl2 + 2 * 24 * 24, tmpA, NE, 0);
  k_onsite<<<gdiv(EW, 256), 256, 0, stream>>>(value, tmpA, edge_vec, NE);

  // 9) edge_msg (vec form) = value * alpha[head(j)]
  k_alpha_scale<<<gdiv(EW, 256), 256, 0, stream>>>(value, alpha, tmpB, NE);

  // 10) node_msg = segment_sum over edge_dst
  k_zero<<<gdiv((long)NN * 360, 256), 256, 0, stream>>>(node_msg, (long)NN * 360);
  k_segsum<<<gdiv(EW, 256), 256, 0, stream>>>(tmpB, edge_dst, node_msg, NE);

  // 11) outputs
  irreps(stream, node_msg, Wl0 + 3 * 96 * 96, Wl1 + 3 * 48 * 48, Wl2 + 3 * 24 * 24, node_out, NN, 0);
  irreps(stream, node_fea, Wl0 + 5 * 96 * 96, Wl1 + 5 * 48 * 48, Wl2 + 5 * 24 * 24, node_out, NN, 1);
  irreps(stream, tmpB,     Wl0 + 4 * 96 * 96, Wl1 + 4 * 48 * 48, Wl2 + 4 * 24 * 24, edge_out, NE, 0);
  irreps(stream, edge_fea, Wl0 + 6 * 96 * 96, Wl1 + 6 * 48 * 48, Wl2 + 6 * 24 * 24, edge_out, NE, 1);
}